// YoloV4_38233798868998
// MI455X (gfx1250) — compile-verified
//
#include <hip/hip_runtime.h>
#include <stdint.h>

// Problem constants (from reference setup_inputs)
#define B_    64
#define N_    4032
#define C_    80
#define KPC   50          // MAX_PER_CLASS
#define KTOT  50          // MAX_TOTAL
#define M_    (C_ * KPC)  // 4000 candidates per image
#define BLK   256
#define ITEMS 16          // 256*16 = 4096 >= 4032

#define SCORE_THR 0.4f
#define IOU_THR   0.45f
#define NEG_SENT  (-3.0e38f)

// ---- CDNA5 TDM availability (amdgpu-toolchain clang-23: 6-arg builtin) ----
#if defined(__has_builtin)
#if __has_builtin(__builtin_amdgcn_tensor_load_to_lds) && \
    __has_builtin(__builtin_amdgcn_s_wait_tensorcnt)
#define USE_TDM 1
#endif
#endif
#ifndef USE_TDM
#define USE_TDM 0
#endif

#if USE_TDM
typedef unsigned int u32x4 __attribute__((ext_vector_type(4)));
typedef int          i32x4 __attribute__((ext_vector_type(4)));
typedef int          i32x8 __attribute__((ext_vector_type(8)));
#endif

// ---------------- wave32 / block argmax (lowest-index tie-break, matches jnp.argmax) ----------------
__device__ __forceinline__ void warpArgmax(float& v, int& i) {
#pragma unroll
  for (int off = 16; off > 0; off >>= 1) {
    float v2 = __shfl_xor(v, off, 32);
    int   i2 = __shfl_xor(i, off, 32);
    if (v2 > v || (v2 == v && i2 < i)) { v = v2; i = i2; }
  }
}

__device__ __forceinline__ void blockArgmax(float& bv, int& bi, float* rv, int* ri) {
  const int tid = threadIdx.x;
  warpArgmax(bv, bi);
  if ((tid & 31) == 0) { rv[tid >> 5] = bv; ri[tid >> 5] = bi; }
  __syncthreads();
  if (tid < 32) {
    float v = (tid < (BLK / 32)) ? rv[tid] : NEG_SENT;
    int   i = (tid < (BLK / 32)) ? ri[tid] : 0x7fffffff;
    warpArgmax(v, i);
    if (tid == 0) { rv[0] = v; ri[0] = i; }
  }
  __syncthreads();
  bv = rv[0]; bi = ri[0];
  __syncthreads();  // rv/ri reused next round
}

// ---------------- Kernel 1: per-(image,class) sequential NMS ----------------
// grid = (C_, B_), block = 256 (8 wave32s)
__global__ __launch_bounds__(BLK) void nms_per_class(
    const float* __restrict__ xywh,    // [B,N,4]
    const float* __restrict__ scores,  // [B,N,C]
    float* __restrict__ candS,         // [B,C,KPC]
    int*   __restrict__ candI)         // [B,C,KPC]
{
  const int c = blockIdx.x;
  const int b = blockIdx.y;
  const int tid = threadIdx.x;

  __shared__ float sy1[N_], sx1[N_], sy2[N_], sx2[N_], sa[N_];  // decoded boxes (SoA)
  __shared__ float sc[N_];                                      // live scores
  __shared__ float rv[BLK / 32];
  __shared__ int   ri[BLK / 32];

#if USE_TDM
  // --- Async strided DMA: scores[b, 0:4032, c] (stride 80 floats) -> LDS sc[] ---
  if ((tid >> 5) == 0) {  // single wave issues (EXEC ignored by TDM; avoid 8x duplicate issue)
    uint64_t ga  = (uint64_t)(uintptr_t)(scores + ((size_t)b * N_) * C_ + c);
    uint32_t lds = (uint32_t)(uintptr_t)(void*)sc;  // generic->LDS: low 32 bits = WG-relative offset
    u32x4 g0 = { 1u,                                  // count=1, no gather/iterate
                 lds,                                 // lds_addr
                 (uint32_t)ga,                        // global_addr[31:0]
                 ((uint32_t)(ga >> 32) & 0x01FFFFFFu) | 0x80000000u };  // addr[56:32] | type=2
    i32x8 g1 = { 0x00020000,          // data_size=4B (code 2), mask=0, no pad/iterate
                 0x00010000,          // tensor_dim0 = 1 (bits [79:48] -> dword1[31:16])
                 (int)(4032u << 16),  // tensor_dim1 = 4032 (dword2[31:16])
                 0x00010000,          // tile_dim0 = 1 (dword3[31:16])
                 0x00000FC0,          // tile_dim1 = 4032, tile_dim2 = 0
                 80,                  // tensor_dim0_stride = 80 elements (class stride)
                 0, 0 };              // tensor_dim1_stride = 0 (unused)
    i32x4 g2 = { 0, 0, 0, 0 };
    i32x4 g3 = { 0, 0, 0, 0 };
    i32x8 g4 = { 0, 0, 0, 0, 0, 0, 0, 0 };  // extra group on clang-23 6-arg form
    __builtin_amdgcn_tensor_load_to_lds(g0, g1, g2, g3, g4, 0);
  }
#endif

  // --- Decode boxes into LDS SoA (L2-hot: reused by all 80 class-blocks of this image) ---
  const float4* bx = (const float4*)(xywh + (size_t)b * N_ * 4);
#pragma unroll
  for (int kk = 0; kk < ITEMS; ++kk) {
    const int j = tid + BLK * kk;
    if (j < N_) {
      float4 r = bx[j];  // x, y, w, h
      const float y1 = (r.y - 0.5f * r.w) * (1.0f / 256.0f);
      const float x1 = (r.x - 0.5f * r.z) * (1.0f / 256.0f);
      const float y2 = (r.y + 0.5f * r.w) * (1.0f / 256.0f);
      const float x2 = (r.x + 0.5f * r.z) * (1.0f / 256.0f);
      sy1[j] = y1; sx1[j] = x1; sy2[j] = y2; sx2[j] = x2;
      sa[j] = fmaxf(y2 - y1, 0.0f) * fmaxf(x2 - x1, 0.0f);
    }
  }

#if USE_TDM
  __builtin_amdgcn_s_wait_tensorcnt((short)0);  // issuing wave drains DMA; others no-op
#else
  // Fallback: direct strided loads
#pragma unroll
  for (int kk = 0; kk < ITEMS; ++kk) {
    const int j = tid + BLK * kk;
    if (j < N_) sc[j] = scores[((size_t)b * N_ + j) * C_ + c];
  }
#endif
  __syncthreads();

  // Score threshold (in-place)
#pragma unroll
  for (int kk = 0; kk < ITEMS; ++kk) {
    const int j = tid + BLK * kk;
    if (j < N_) {
      const float s = sc[j];
      sc[j] = (s >= SCORE_THR) ? s : -1.0f;
    }
  }
  __syncthreads();

  float* outS = candS + ((size_t)b * C_ + c) * KPC;
  int*   outI = candI + ((size_t)b * C_ + c) * KPC;

  for (int k = 0; k < KPC; ++k) {
    // block argmax over live scores
    float bv = NEG_SENT; int bi = 0x7fffffff;
#pragma unroll
    for (int kk = 0; kk < ITEMS; ++kk) {
      const int j = tid + BLK * kk;
      if (j < N_) {
        const float s = sc[j];
        if (s > bv) { bv = s; bi = j; }
      }
    }
    blockArgmax(bv, bi, rv, ri);

    const bool valid = bv > 0.0f;  // uniform across block
    if (!valid) {
      if (tid == 0) {
        for (int k2 = k; k2 < KPC; ++k2) { outS[k2] = -1.0f; outI[k2] = 0; }
      }
      break;  // once invalid, all later rounds invalid too
    }
    if (tid == 0) { outS[k] = bv; outI[k] = bi; }

    // Suppress by IoU vs winner (winner self-suppresses: IoU ~ 1 > thr)
    const float by1 = sy1[bi], bxx1 = sx1[bi], by2 = sy2[bi], bxx2 = sx2[bi], ba = sa[bi];
#pragma unroll
    for (int kk = 0; kk < ITEMS; ++kk) {
      const int j = tid + BLK * kk;
      if (j < N_) {
        const float s = sc[j];
        if (s > -0.5f) {  // only live candidates
          const float yA = fmaxf(by1, sy1[j]);
          const float xA = fmaxf(bxx1, sx1[j]);
          const float yB = fminf(by2, sy2[j]);
          const float xB = fminf(bxx2, sx2[j]);
          const float inter = fmaxf(yB - yA, 0.0f) * fmaxf(xB - xA, 0.0f);
          const float iou = inter / (ba + sa[j] - inter + 1e-9f);
          if (iou > IOU_THR) sc[j] = -1.0f;
        }
      }
    }
    __syncthreads();
  }
}

// ---------------- Kernel 2: per-image top-50 over 4000 candidates + output assembly ----------------
// grid = B_, block = 256
__global__ __launch_bounds__(BLK) void topk_finalize(
    const float* __restrict__ xywh,   // [B,N,4]
    const float* __restrict__ candS,  // [B,C,KPC]
    const int*   __restrict__ candI,  // [B,C,KPC]
    float* __restrict__ out)          // boxes | scores | classes | n_valid(int bits)
{
  const int b = blockIdx.x;
  const int tid = threadIdx.x;

  __shared__ float s[BLK * ITEMS];  // 4096 slots
  __shared__ float rv[BLK / 32];
  __shared__ int   ri[BLK / 32];

#pragma unroll
  for (int kk = 0; kk < ITEMS; ++kk) {
    const int j = tid + BLK * kk;
    s[j] = (j < M_) ? candS[(size_t)b * M_ + j] : -2.0f;  // -2 < any candidate (-1 = invalid)
  }
  __syncthreads();

  float* ob  = out + (size_t)b * KTOT * 4;
  float* os  = out + (size_t)B_ * KTOT * 4 + (size_t)b * KTOT;
  float* oc  = out + (size_t)B_ * KTOT * 4 + (size_t)B_ * KTOT + (size_t)b * KTOT;
  int*   onv = (int*)(out + (size_t)B_ * KTOT * 4 + 2 * (size_t)B_ * KTOT) + b;

  int nv = 0;
  for (int k = 0; k < KTOT; ++k) {
    float bv = NEG_SENT; int bi = 0x7fffffff;
#pragma unroll
    for (int kk = 0; kk < ITEMS; ++kk) {
      const int j = tid + BLK * kk;
      const float v = s[j];
      if (v > bv) { bv = v; bi = j; }
    }
    blockArgmax(bv, bi, rv, ri);

    if (tid == 0) {
      const bool keep = bv > 0.0f;
      if (keep) nv++;
      const int cls  = bi / KPC;
      const int bidx = candI[(size_t)b * M_ + bi];
      const float* r = xywh + ((size_t)b * N_ + (size_t)bidx) * 4;
      const float x = r[0], y = r[1], w = r[2], h = r[3];
      float y1 = fminf(fmaxf((y - 0.5f * h) * (1.0f / 256.0f), 0.0f), 1.0f);
      float x1 = fminf(fmaxf((x - 0.5f * w) * (1.0f / 256.0f), 0.0f), 1.0f);
      float y2 = fminf(fmaxf((y + 0.5f * h) * (1.0f / 256.0f), 0.0f), 1.0f);
      float x2 = fminf(fmaxf((x + 0.5f * w) * (1.0f / 256.0f), 0.0f), 1.0f);
      ob[k * 4 + 0] = keep ? y1 : 0.0f;
      ob[k * 4 + 1] = keep ? x1 : 0.0f;
      ob[k * 4 + 2] = keep ? y2 : 0.0f;
      ob[k * 4 + 3] = keep ? x2 : 0.0f;
      os[k] = keep ? bv : 0.0f;
      oc[k] = keep ? (float)cls : 0.0f;
      s[bi] = -2.0f;  // remove from pool (distinct picks, like top_k)
    }
    __syncthreads();
  }
  if (tid == 0) *onv = nv;  // int32 bits, matching reference's int32 valid_detections
}

// ---------------- launch ----------------
extern "C" void kernel_launch(void* const* d_in, const int* in_sizes, int n_in,
                              void* d_out, int out_size, void* d_ws, size_t ws_size,
                              hipStream_t stream) {
  const float* xywh   = (const float*)d_in[0];  // [64,4032,4] f32
  const float* scores = (const float*)d_in[1];  // [64,4032,80] f32

  // workspace: candidate scores then candidate indices (~2 MB total)
  float* candS = (float*)d_ws;
  int*   candI = (int*)((char*)d_ws + (size_t)B_ * C_ * KPC * sizeof(float));

  nms_per_class<<<dim3(C_, B_), BLK, 0, stream>>>(xywh, scores, candS, candI);
  topk_finalize<<<B_, BLK, 0, stream>>>(xywh, candS, candI, (float*)d_out);
}